// Net_61375082659915
// MI455X (gfx1250) — compile-verified
//
#include <hip/hip_runtime.h>

typedef __attribute__((ext_vector_type(2))) float v2f;
typedef __attribute__((ext_vector_type(8))) float v8f;

#define NN 100000
#define NE 1600000
#define NG 1000

// ---------------- elementwise copy ----------------
__global__ void copy_f32(const float* __restrict__ in, float* __restrict__ out, long long n) {
  long long i = (long long)blockIdx.x * blockDim.x + threadIdx.x;
  if (i < n) out[i] = in[i];
}

// ---------------- GIN edge scatter: agg[dst] += h[src] ----------------
__global__ void edge_scatter(const float* __restrict__ h, const int* __restrict__ src,
                             const int* __restrict__ dst, float* __restrict__ agg, int F) {
  long long i = (long long)blockIdx.x * blockDim.x + threadIdx.x;
  long long total = (long long)NE * F;
  if (i >= total) return;
  int e = (int)(i / F);
  int f = (int)(i - (long long)e * F);
  int s = src[e], d = dst[e];
  atomicAdd(&agg[(long long)d * F + f], h[(long long)s * F + f]);
}

// ---------------- GEMM + bias via fp32 WMMA 16x16x4, LDS-staged ----------------
// Block = colTiles waves; each wave owns one 16-col tile; block shares a 16-row
// X stripe staged in LDS; W staged transposed+zero-padded in LDS. Inner loop is
// branch-free: 2x ds_load_b64 + v_wmma_f32_16x16x4_f32 per K-step.
// Fragment layouts per CDNA5 ISA 7.12.2:
//  A 16x4 f32: lane%16 = row M; lane/16 selects K-pair; vgpr0=K, vgpr1=K+1
//  B 4x16 f32: lane%16 = col N; lane/16 selects K-pair; vgpr0=K, vgpr1=K+1
//  C/D 16x16:  lane%16 = col; vgpr v -> row v + 8*(lane/16)
template <int Fin, int Fout>
__global__ void gemm_bias_wmma(const float* __restrict__ X, const float* __restrict__ W,
                               const float* __restrict__ bias, float* __restrict__ Y) {
  constexpr int Fpad = (Fin + 3) & ~3;
  constexpr int CT   = (Fout + 15) / 16;
  constexpr int NTHR = CT * 32;
  __shared__ float lsX[16 * Fpad];        // [row][k]
  __shared__ float lsW[CT * 16 * Fpad];   // transposed: [col][k], zero-padded

  const int rowBase = blockIdx.x * 16;
  const int tid = threadIdx.x;

  for (int idx = tid; idx < 16 * Fpad; idx += NTHR) {
    int r = idx / Fpad, k = idx - r * Fpad;
    lsX[idx] = (k < Fin) ? X[(long long)(rowBase + r) * Fin + k] : 0.0f;
  }
  for (int idx = tid; idx < CT * 16 * Fpad; idx += NTHR) {
    int c = idx / Fpad, k = idx - c * Fpad;
    lsW[idx] = (k < Fin && c < Fout) ? W[(long long)k * Fout + c] : 0.0f;
  }
  __syncthreads();

  const int wave  = tid >> 5;      // column tile
  const int lane  = tid & 31;
  const int l16   = lane & 15;
  const int khalf = lane >> 4;     // 0 or 1
  const int col   = wave * 16 + l16;

  v8f acc = {};
  const float* ax = &lsX[l16 * Fpad + 2 * khalf];
  const float* bw = &lsW[col * Fpad + 2 * khalf];
#pragma unroll
  for (int k = 0; k < Fpad; k += 4) {
    v2f a, b;
    a.x = ax[k];     a.y = ax[k + 1];
    b.x = bw[k];     b.y = bw[k + 1];
    acc = __builtin_amdgcn_wmma_f32_16x16x4_f32(false, a, false, b, (short)0, acc,
                                                false, false);
  }

  if (col < Fout) {
    float bv = bias[col];
#pragma unroll
    for (int v = 0; v < 8; ++v) {
      int row = rowBase + v + 8 * khalf;
      Y[(long long)row * Fout + col] = acc[v] + bv;
    }
  }
}

// ---------------- BN column reduction: sums[0..F)=sum, sums[F..2F)=sumsq ----------------
__global__ void bn_reduce(const float* __restrict__ Y, int F, float* __restrict__ sums) {
  __shared__ float ls[256];
  for (int t = threadIdx.x; t < 2 * F; t += blockDim.x) ls[t] = 0.0f;
  __syncthreads();
  long long total  = (long long)NN * F;
  long long stride = (long long)gridDim.x * blockDim.x;
  for (long long i = (long long)blockIdx.x * blockDim.x + threadIdx.x; i < total; i += stride) {
    int f = (int)(i % F);
    float v = Y[i];
    atomicAdd(&ls[f], v);
    atomicAdd(&ls[F + f], v * v);
  }
  __syncthreads();
  for (int t = threadIdx.x; t < 2 * F; t += blockDim.x) atomicAdd(&sums[t], ls[t]);
}

__global__ void bn_finalize(const float* __restrict__ sums, const float* __restrict__ gamma,
                            const float* __restrict__ beta, int F, float invN,
                            float* __restrict__ ss) {
  int f = blockIdx.x * blockDim.x + threadIdx.x;
  if (f >= F) return;
  float mean  = sums[f] * invN;
  float var   = sums[F + f] * invN - mean * mean;
  float scale = gamma[f] * rsqrtf(var + 1e-5f);
  ss[f]     = scale;
  ss[F + f] = beta[f] - mean * scale;
}

__global__ void bn_apply_relu(float* __restrict__ Y, const float* __restrict__ ss, int F) {
  long long i = (long long)blockIdx.x * blockDim.x + threadIdx.x;
  long long total = (long long)NN * F;
  if (i >= total) return;
  int f = (int)(i % F);
  float v = Y[i] * ss[f] + ss[F + f];
  Y[i] = v > 0.0f ? v : 0.0f;
}

// ---------------- per-graph mean readout accumulation ----------------
__global__ void graph_reduce(const float* __restrict__ h, const int* __restrict__ gid,
                             float* __restrict__ gsum, float* __restrict__ gcnt) {
  long long i = (long long)blockIdx.x * blockDim.x + threadIdx.x;
  if (i >= (long long)NN * 20) return;
  int node = (int)(i / 20);
  int f    = (int)(i - (long long)node * 20);
  int g = gid[node];
  atomicAdd(&gsum[(long long)g * 20 + f], h[i]);
  if (f == 0) atomicAdd(&gcnt[g], 1.0f);
}

// ---------------- head MLP: concat(mean, self_feat) -> 10 -> 1 ----------------
__global__ void head_mlp(const float* __restrict__ gsum, const float* __restrict__ gcnt,
                         const float* __restrict__ self_feat,
                         const float* __restrict__ w1, const float* __restrict__ b1,
                         const float* __restrict__ w2, const float* __restrict__ b2,
                         float* __restrict__ out) {
  int g = blockIdx.x * blockDim.x + threadIdx.x;
  if (g >= NG) return;
  float hg[23];
  float c = gcnt[g];
  c = (c > 1.0f) ? c : 1.0f;
#pragma unroll
  for (int f = 0; f < 20; ++f) hg[f] = gsum[g * 20 + f] / c;
#pragma unroll
  for (int f = 0; f < 3; ++f) hg[20 + f] = self_feat[g * 3 + f];
  float o = b2[0];
#pragma unroll
  for (int j = 0; j < 10; ++j) {
    float s = b1[j];
#pragma unroll
    for (int k = 0; k < 23; ++k) s += hg[k] * w1[k * 10 + j];
    s = (s > 0.0f) ? s : 0.0f;
    o += s * w2[j];
  }
  out[g] = o;
}

// ---------------- orchestration ----------------
static inline long long cdiv(long long a, long long b) { return (a + b - 1) / b; }

extern "C" void kernel_launch(void* const* d_in, const int* in_sizes, int n_in,
                              void* d_out, int out_size, void* d_ws, size_t ws_size,
                              hipStream_t stream) {
  const float* feat      = (const float*)d_in[0];
  const float* self_feat = (const float*)d_in[1];
  const int*   src       = (const int*)d_in[2];
  const int*   dst       = (const int*)d_in[3];
  const int*   gid       = (const int*)d_in[4];
  const float* w1a = (const float*)d_in[5],  *b1a = (const float*)d_in[6];
  const float* g1a = (const float*)d_in[7],  *be1a = (const float*)d_in[8];
  const float* w1b = (const float*)d_in[9],  *b1b = (const float*)d_in[10];
  const float* gbn1 = (const float*)d_in[11], *bbn1 = (const float*)d_in[12];
  const float* w2a = (const float*)d_in[13], *b2a = (const float*)d_in[14];
  const float* g2a = (const float*)d_in[15], *be2a = (const float*)d_in[16];
  const float* w2b = (const float*)d_in[17], *b2b = (const float*)d_in[18];
  const float* gbn2 = (const float*)d_in[19], *bbn2 = (const float*)d_in[20];
  const float* wfc1 = (const float*)d_in[21], *bfc1 = (const float*)d_in[22];
  const float* wfc2 = (const float*)d_in[23], *bfc2 = (const float*)d_in[24];
  float* out = (float*)d_out;

  float* ws    = (float*)d_ws;
  float* buf0  = ws;                       // [NN,100] max
  float* buf1  = ws + 10000000LL;          // [NN,100]
  float* buf2  = ws + 20000000LL;          // [NN,100]
  float* bnsum = ws + 30000000LL;          // 256
  float* bnss  = bnsum + 256;              // 256
  float* gsum  = bnss + 256;               // 20000
  float* gcnt  = gsum + 20000;             // 1000

  const int B = 256;
  const int rowTiles = NN / 16;            // 6250

  auto run_bn = [&](float* Y, int F, const float* gamma, const float* beta) {
    hipMemsetAsync(bnsum, 0, 2 * F * sizeof(float), stream);
    bn_reduce<<<2048, B, 0, stream>>>(Y, F, bnsum);
    bn_finalize<<<1, 128, 0, stream>>>(bnsum, gamma, beta, F, 1.0f / (float)NN, bnss);
    bn_apply_relu<<<(int)cdiv((long long)NN * F, B), B, 0, stream>>>(Y, bnss, F);
  };

  // ---- GIN layer 1: agg = feat + scatter(feat) ----
  copy_f32<<<(int)cdiv((long long)NN * 50, B), B, 0, stream>>>(feat, buf0, (long long)NN * 50);
  edge_scatter<<<(int)cdiv((long long)NE * 50, B), B, 0, stream>>>(feat, src, dst, buf0, 50);
  // mlp: [N,50]@[50,100] -> BN/ReLU -> [N,100]@[100,100] -> BN/ReLU
  gemm_bias_wmma<50, 100><<<rowTiles, 7 * 32, 0, stream>>>(buf0, w1a, b1a, buf1);
  run_bn(buf1, 100, g1a, be1a);
  gemm_bias_wmma<100, 100><<<rowTiles, 7 * 32, 0, stream>>>(buf1, w1b, b1b, buf2);
  run_bn(buf2, 100, gbn1, bbn1);

  // ---- GIN layer 2: agg = h + scatter(h) ----
  copy_f32<<<(int)cdiv((long long)NN * 100, B), B, 0, stream>>>(buf2, buf0, (long long)NN * 100);
  edge_scatter<<<(int)cdiv((long long)NE * 100, B), B, 0, stream>>>(buf2, src, dst, buf0, 100);
  // mlp: [N,100]@[100,20] -> BN/ReLU -> [N,20]@[20,20] -> BN/ReLU
  gemm_bias_wmma<100, 20><<<rowTiles, 2 * 32, 0, stream>>>(buf0, w2a, b2a, buf1);
  run_bn(buf1, 20, g2a, be2a);
  gemm_bias_wmma<20, 20><<<rowTiles, 2 * 32, 0, stream>>>(buf1, w2b, b2b, buf2);
  run_bn(buf2, 20, gbn2, bbn2);

  // ---- mean readout + head ----
  hipMemsetAsync(gsum, 0, (20000 + 1000) * sizeof(float), stream);
  graph_reduce<<<(int)cdiv((long long)NN * 20, B), B, 0, stream>>>(buf2, gid, gsum, gcnt);
  head_mlp<<<(int)cdiv(NG, B), B, 0, stream>>>(gsum, gcnt, self_feat,
                                               wfc1, bfc1, wfc2, bfc2, out);
}